// Net_44281112821760
// MI455X (gfx1250) — compile-verified
//
#include <hip/hip_runtime.h>
#include <math.h>

typedef __attribute__((ext_vector_type(2))) float v2f;
typedef __attribute__((ext_vector_type(8))) float v8f;

#define F_IN 256
#define H_DIM 16
#define C_DIM 32

// ---------------------------------------------------------------- utilities

__global__ void zero_f32_kernel(float* __restrict__ p, size_t n) {
    size_t i = (size_t)blockIdx.x * blockDim.x + threadIdx.x;
    size_t stride = (size_t)gridDim.x * blockDim.x;
    for (; i < n; i += stride) p[i] = 0.0f;
}

// out[dst[e]] += h[src[e]] for 256-wide rows; 64 lanes per edge, float4 gathers.
__global__ void scatter_add_256_kernel(const float* __restrict__ h,
                                       const int* __restrict__ src,
                                       const int* __restrict__ dst,
                                       float* __restrict__ out, int E) {
    unsigned idx = blockIdx.x * 256u + threadIdx.x;
    int e = idx >> 6;
    if (e >= E) return;
    int lane = idx & 63;
    int s = src[e], d = dst[e];
    const float4* hs = (const float4*)(h + (size_t)s * F_IN);
    float4 v = hs[lane];
    float* o = out + (size_t)d * F_IN + lane * 4;
    unsafeAtomicAdd(o + 0, v.x);
    unsafeAtomicAdd(o + 1, v.y);
    unsafeAtomicAdd(o + 2, v.z);
    unsafeAtomicAdd(o + 3, v.w);
}

// out[dst[e]] += h[src[e]] for 16-wide rows; 16 lanes per edge.
__global__ void scatter_add_16_kernel(const float* __restrict__ h,
                                      const int* __restrict__ src,
                                      const int* __restrict__ dst,
                                      float* __restrict__ out, int E) {
    unsigned idx = blockIdx.x * 256u + threadIdx.x;
    int e = idx >> 4;
    if (e >= E) return;
    int lane = idx & 15;
    int s = src[e], d = dst[e];
    float v = h[(size_t)s * H_DIM + lane];
    unsafeAtomicAdd(out + (size_t)d * H_DIM + lane, v);
}

// ---------------------------------------------------------------- GEMM 1
// h1 = x + (x + invK*aggr) @ W1inter + b1inter      (N,256)x(256,256)
// block: 512 threads = 16 waves; block handles 16 rows x 256 cols.
__global__ void __launch_bounds__(512) gemm1_fused_kernel(
    const float* __restrict__ x, const float* __restrict__ aggr,
    const float* __restrict__ W, const float* __restrict__ bias,
    const int* __restrict__ Kp, float* __restrict__ out) {
    __shared__ float u[16][F_IN + 4];  // stride 260 -> conflict-free col reads
    const float invK = 1.0f / (float)(*Kp);
    const size_t rowBase = (size_t)blockIdx.x * 16;

    for (int i = threadIdx.x; i < 16 * F_IN; i += 512) {
        int r = i >> 8, c = i & (F_IN - 1);
        size_t g = (rowBase + r) * F_IN + c;
        u[r][c] = x[g] + invK * aggr[g];
    }
    __syncthreads();

    const int wave = threadIdx.x >> 5;
    const int lane = threadIdx.x & 31;
    const int colg = (wave << 4) + (lane & 15);   // 0..255
    const int arow = lane & 15;
    const int koff = (lane >> 4) << 1;            // 0 or 2

    v8f acc = {};
    for (int k = 0; k < F_IN; k += 4) {
        v2f a, b;
        a[0] = u[arow][k + koff];
        a[1] = u[arow][k + koff + 1];
        b[0] = W[(size_t)(k + koff) * F_IN + colg];
        b[1] = W[(size_t)(k + koff + 1) * F_IN + colg];
        acc = __builtin_amdgcn_wmma_f32_16x16x4_f32(
            false, a, false, b, (short)0, acc, false, false);
    }

    const float bc = bias[colg];
    const int rbase = (lane >> 4) << 3;
#pragma unroll
    for (int v = 0; v < 8; ++v) {
        size_t g = (rowBase + rbase + v) * F_IN + colg;
        out[g] = x[g] + acc[v] + bc;
    }
}

// ---------------------------------------------------------------- GEMM 2
// h2 = relu((h1 + aggr) @ W1intra + b1intra)        (N,256)x(256,16)
__global__ void __launch_bounds__(32) gemm2_fused_kernel(
    const float* __restrict__ h1, const float* __restrict__ aggr,
    const float* __restrict__ W, const float* __restrict__ bias,
    float* __restrict__ out) {
    const size_t rowBase = (size_t)blockIdx.x * 16;
    const int lane = threadIdx.x & 31;
    const int col = lane & 15;
    const int arow = lane & 15;
    const int koff = (lane >> 4) << 1;

    v8f acc = {};
    for (int k = 0; k < F_IN; k += 4) {
        v2f a, b;
        size_t g0 = (rowBase + arow) * F_IN + k + koff;
        a[0] = h1[g0] + aggr[g0];
        a[1] = h1[g0 + 1] + aggr[g0 + 1];
        b[0] = W[(size_t)(k + koff) * H_DIM + col];
        b[1] = W[(size_t)(k + koff + 1) * H_DIM + col];
        acc = __builtin_amdgcn_wmma_f32_16x16x4_f32(
            false, a, false, b, (short)0, acc, false, false);
    }

    const float bc = bias[col];
    const int rbase = (lane >> 4) << 3;
#pragma unroll
    for (int v = 0; v < 8; ++v) {
        float t = acc[v] + bc;
        out[(rowBase + rbase + v) * H_DIM + col] = t > 0.0f ? t : 0.0f;
    }
}

// ---------------------------------------------------------------- GEMM 3
// h3 = h2 + (h2 + invK*aggr) @ W2inter + b2inter    (N,16)x(16,16)
__global__ void __launch_bounds__(32) gemm3_fused_kernel(
    const float* __restrict__ h2, const float* __restrict__ aggr,
    const float* __restrict__ W, const float* __restrict__ bias,
    const int* __restrict__ Kp, float* __restrict__ out) {
    const float invK = 1.0f / (float)(*Kp);
    const size_t rowBase = (size_t)blockIdx.x * 16;
    const int lane = threadIdx.x & 31;
    const int col = lane & 15;
    const int arow = lane & 15;
    const int koff = (lane >> 4) << 1;

    v8f acc = {};
#pragma unroll
    for (int k = 0; k < H_DIM; k += 4) {
        v2f a, b;
        size_t g0 = (rowBase + arow) * H_DIM + k + koff;
        a[0] = h2[g0] + invK * aggr[g0];
        a[1] = h2[g0 + 1] + invK * aggr[g0 + 1];
        b[0] = W[(size_t)(k + koff) * H_DIM + col];
        b[1] = W[(size_t)(k + koff + 1) * H_DIM + col];
        acc = __builtin_amdgcn_wmma_f32_16x16x4_f32(
            false, a, false, b, (short)0, acc, false, false);
    }

    const float bc = bias[col];
    const int rbase = (lane >> 4) << 3;
#pragma unroll
    for (int v = 0; v < 8; ++v) {
        size_t g = (rowBase + rbase + v) * H_DIM + col;
        out[g] = h2[g] + acc[v] + bc;
    }
}

// ---------------------------------------------------------------- GEMM 4
// logits = (h3 + aggr) @ W2intra + b2intra          (N,16)x(16,32)
__global__ void __launch_bounds__(32) gemm4_fused_kernel(
    const float* __restrict__ h3, const float* __restrict__ aggr,
    const float* __restrict__ W, const float* __restrict__ bias,
    float* __restrict__ out) {
    const size_t rowBase = (size_t)blockIdx.x * 16;
    const int lane = threadIdx.x & 31;
    const int col = lane & 15;
    const int arow = lane & 15;
    const int koff = (lane >> 4) << 1;

    v8f accA = {}, accB = {};
#pragma unroll
    for (int k = 0; k < H_DIM; k += 4) {
        v2f a, bA, bB;
        size_t g0 = (rowBase + arow) * H_DIM + k + koff;
        a[0] = h3[g0] + aggr[g0];
        a[1] = h3[g0 + 1] + aggr[g0 + 1];
        bA[0] = W[(size_t)(k + koff) * C_DIM + col];
        bA[1] = W[(size_t)(k + koff + 1) * C_DIM + col];
        bB[0] = W[(size_t)(k + koff) * C_DIM + col + 16];
        bB[1] = W[(size_t)(k + koff + 1) * C_DIM + col + 16];
        accA = __builtin_amdgcn_wmma_f32_16x16x4_f32(
            false, a, false, bA, (short)0, accA, false, false);
        accB = __builtin_amdgcn_wmma_f32_16x16x4_f32(
            false, a, false, bB, (short)0, accB, false, false);
    }

    const float bcA = bias[col];
    const float bcB = bias[col + 16];
    const int rbase = (lane >> 4) << 3;
#pragma unroll
    for (int v = 0; v < 8; ++v) {
        size_t g = (rowBase + rbase + v) * C_DIM + col;
        out[g] = accA[v] + bcA;
        out[g + 16] = accB[v] + bcB;
    }
}

// ---------------------------------------------------------------- softmax
__global__ void log_softmax_kernel(const float* __restrict__ logits,
                                   float* __restrict__ out, int N) {
    int r = blockIdx.x * blockDim.x + threadIdx.x;
    if (r >= N) return;
    const float* p = logits + (size_t)r * C_DIM;
    float m = p[0];
#pragma unroll
    for (int i = 1; i < C_DIM; ++i) m = fmaxf(m, p[i]);
    float s = 0.0f;
#pragma unroll
    for (int i = 0; i < C_DIM; ++i) s += expf(p[i] - m);
    float ls = logf(s);
    float* o = out + (size_t)r * C_DIM;
#pragma unroll
    for (int i = 0; i < C_DIM; ++i) o[i] = p[i] - m - ls;
}

// ---------------------------------------------------------------- driver

extern "C" void kernel_launch(void* const* d_in, const int* in_sizes, int n_in,
                              void* d_out, int out_size, void* d_ws, size_t ws_size,
                              hipStream_t stream) {
    const float* x       = (const float*)d_in[0];
    const int*   edge    = (const int*)d_in[1];
    const int*   ego     = (const int*)d_in[2];
    const float* W1inter = (const float*)d_in[3];
    const float* b1inter = (const float*)d_in[4];
    const float* W1intra = (const float*)d_in[5];
    const float* b1intra = (const float*)d_in[6];
    const float* W2inter = (const float*)d_in[7];
    const float* b2inter = (const float*)d_in[8];
    const float* W2intra = (const float*)d_in[9];
    const float* b2intra = (const float*)d_in[10];
    const int*   Kp      = (const int*)d_in[11];

    const int N = in_sizes[0] / F_IN;
    const int E = in_sizes[1] / 2;
    const int* src_m = edge;
    const int* dst_m = edge + E;
    const int* src_e = ego;
    const int* dst_e = ego + E;

    float* ws = (float*)d_ws;
    const size_t nf256 = (size_t)N * F_IN;
    const size_t nf16  = (size_t)N * H_DIM;
    float* aggr256 = ws;                 // N x 256 (reused ego/main)
    float* h1      = ws + nf256;         // N x 256
    float* h2      = ws + 2 * nf256;     // N x 16
    float* aggr16  = h2 + nf16;          // N x 16 (reused ego/main)
    float* h3      = aggr16 + nf16;      // N x 16
    float* logits  = h3 + nf16;          // N x 32
    float* out     = (float*)d_out;

    const int tiles = N / 16;                                  // 6250
    const unsigned g_sc256 = ((unsigned)E * 64u + 255u) / 256u;
    const unsigned g_sc16  = ((unsigned)E * 16u + 255u) / 256u;

    // ---- layer 1: inter (ego) conv, 256-dim
    zero_f32_kernel<<<2048, 256, 0, stream>>>(aggr256, nf256);
    scatter_add_256_kernel<<<g_sc256, 256, 0, stream>>>(x, src_e, dst_e, aggr256, E);
    gemm1_fused_kernel<<<tiles, 512, 0, stream>>>(x, aggr256, W1inter, b1inter, Kp, h1);

    // ---- layer 1: intra (main) GIN, 256 -> 16, relu
    zero_f32_kernel<<<2048, 256, 0, stream>>>(aggr256, nf256);
    scatter_add_256_kernel<<<g_sc256, 256, 0, stream>>>(h1, src_m, dst_m, aggr256, E);
    gemm2_fused_kernel<<<tiles, 32, 0, stream>>>(h1, aggr256, W1intra, b1intra, h2);

    // ---- layer 2: inter (ego) conv, 16-dim
    zero_f32_kernel<<<512, 256, 0, stream>>>(aggr16, nf16);
    scatter_add_16_kernel<<<g_sc16, 256, 0, stream>>>(h2, src_e, dst_e, aggr16, E);
    gemm3_fused_kernel<<<tiles, 32, 0, stream>>>(h2, aggr16, W2inter, b2inter, Kp, h3);

    // ---- layer 2: intra (main) GIN, 16 -> 32
    zero_f32_kernel<<<512, 256, 0, stream>>>(aggr16, nf16);
    scatter_add_16_kernel<<<g_sc16, 256, 0, stream>>>(h3, src_m, dst_m, aggr16, E);
    gemm4_fused_kernel<<<tiles, 32, 0, stream>>>(h3, aggr16, W2intra, b2intra, logits);

    // ---- log_softmax
    log_softmax_kernel<<<(N + 255) / 256, 256, 0, stream>>>(logits, out, N);
}